// ModelCLPM_72335839199818
// MI455X (gfx1250) — compile-verified
//
#include <hip/hip_runtime.h>
#include <math.h>

typedef __attribute__((ext_vector_type(2))) float v2f;
typedef __attribute__((ext_vector_type(8))) float v8f;

#define KSEG     8
#define NPK      9        // K+1
#define ZSTRIDE  18       // 2*(K+1) floats per node
#define N_NODES_C 8192
#define PENALTY_F 10.0f

// ---------------- fast device math (native TRANS ops) ----------------
__device__ __forceinline__ float fexp(float x) {
    return __builtin_amdgcn_exp2f(x * 1.44269504088896340736f);
}

// Phi(x) = 0.5 * erfc(-x / sqrt(2)); erfc via Abramowitz-Stegun 7.1.26
__device__ __forceinline__ float fnormcdf(float x) {
    float z  = -0.70710678118654752f * x;   // erfc argument
    float az = fabsf(z);
    float t  = __builtin_amdgcn_rcpf(fmaf(0.3275911f, az, 1.0f));
    float p  = fmaf(t, 1.061405429f, -1.453152027f);
    p        = fmaf(t, p, 1.421413741f);
    p        = fmaf(t, p, -0.284496736f);
    p        = fmaf(t, p, 0.254829592f);
    p       *= t;
    float ec = p * fexp(-az * az);          // erfc(az), az >= 0
    ec       = (z < 0.f) ? (2.f - ec) : ec; // reflect
    return 0.5f * ec;
}

// ---------------- workspace zeroing ----------------
__global__ void clpm_zero_ws(float* ws) {
    ws[0] = 0.f;  // prior raw sum
    ws[1] = 0.f;  // flt sum
    ws[2] = 0.f;  // integral (already * 0.5 * dt)
}

__device__ __forceinline__ float clpm_block_reduce(float v, float* red) {
    int tid = threadIdx.x;
    red[tid] = v;
    __syncthreads();
    for (int s = 128; s > 0; s >>= 1) {
        if (tid < s) red[tid] += red[tid + s];
        __syncthreads();
    }
    return red[0];
}

// ---------------- prior term ----------------
__global__ void clpm_prior_kernel(const float* __restrict__ Z,
                                  const int* __restrict__ nodes,
                                  int bs, float* ws) {
    __shared__ float red[256];
    int g = blockIdx.x * 256 + threadIdx.x;
    float acc = 0.f;
    if (g < bs) {
        const float* zp = Z + (long)nodes[g] * ZSTRIDE;
        float a = zp[0] * zp[0] + zp[NPK] * zp[NPK];
        #pragma unroll
        for (int k = 1; k < NPK; ++k) {
            float dx = zp[k] - zp[k - 1];
            float dy = zp[NPK + k] - zp[NPK + k - 1];
            a += dx * dx + dy * dy;
        }
        acc = a;
    }
    float s = clpm_block_reduce(acc, red);
    if (threadIdx.x == 0) atomicAdd(ws + 0, s);
}

// ---------------- flt term (per-edge) ----------------
__global__ void clpm_flt_kernel(const float* __restrict__ Z,
                                const float* __restrict__ ts,
                                const int* __restrict__ snd,
                                const int* __restrict__ rcv,
                                const float* __restrict__ cp,
                                int E, float* ws) {
    __shared__ float red[256];
    float seg = cp[1] - cp[0];
    float inv_seg = 1.0f / seg;
    float acc = 0.f;
    for (int e = blockIdx.x * 256 + threadIdx.x; e < E; e += gridDim.x * 256) {
        float t = ts[e];
        float q = t * inv_seg;
        int kap = (int)floorf(q);
        kap = kap < 0 ? 0 : (kap > KSEG - 1 ? KSEG - 1 : kap);
        float d  = q - (float)kap;
        float od = 1.f - d;
        const float* zs = Z + (long)snd[e] * ZSTRIDE;
        const float* zr = Z + (long)rcv[e] * ZSTRIDE;
        float scx = zs[kap],     scy = zs[NPK + kap];
        float snx = zs[kap + 1], sny = zs[NPK + kap + 1];
        float rcx = zr[kap],     rcy = zr[NPK + kap];
        float rnx = zr[kap + 1], rny = zr[NPK + kap + 1];
        float cc  = scx * rcx + scy * rcy;   // Zs_c . Zr_c
        float ssq = scx * scx + scy * scy;   // |Zs_c|^2
        float rsq = rcx * rcx + rcy * rcy;   // |Zr_c|^2
        float cn  = scx * rnx + scy * rny;   // Zs_c . Zr_n
        float nc  = snx * rcx + sny * rcy;   // Zs_n . Zr_c
        float sn  = scx * snx + scy * sny;   // Zs_c . Zs_n
        float rn  = rcx * rnx + rcy * rny;   // Zr_c . Zr_n
        float nn  = snx * rnx + sny * rny;   // Zs_n . Zr_n
        float s2  = snx * snx + sny * sny;   // |Zs_n|^2
        float r2  = rnx * rnx + rny * rny;   // |Zr_n|^2
        acc += od * od * (2.f * cc - ssq - rsq)
             + 2.f * d * od * (cn + nc - sn - rn)
             + d * d * (2.f * nn - s2 - r2);
    }
    float s = clpm_block_reduce(acc, red);
    if (threadIdx.x == 0) atomicAdd(ws + 1, s);
}

// ---------------- integral term (WMMA 16x16x4 f32) ----------------
// block = 256 thr (8 waves). blockIdx: [segment k (8)] x [sender tile (128)].
// Each wave sweeps receiver tiles jt = wave, wave+8, ... (512/8 = 64 tiles).
__global__ void __launch_bounds__(256)
clpm_integral_kernel(const float* __restrict__ Z,
                     const int* __restrict__ su,
                     const float* __restrict__ cp,
                     float* ws) {
    __shared__ float sS0x[16], sS0y[16], sDsx[16], sDsy[16];
    __shared__ float sNs0[16], sNds[16], sCs[16];
    __shared__ float red[256];

    int b     = blockIdx.x;
    int k     = b >> 7;        // segment 0..7
    int itile = b & 127;       // sender tile 0..127
    int tid   = threadIdx.x;
    int lane  = tid & 31;
    int hi    = (lane >> 4) & 1;
    int mn    = lane & 15;
    int wave  = tid >> 5;

    if (tid < 16) {
        int sidx = su[itile * 16 + tid];
        const float* zp = Z + (long)sidx * ZSTRIDE;
        float s0x = zp[k],        s1x = zp[k + 1];
        float s0y = zp[NPK + k],  s1y = zp[NPK + k + 1];
        float dsx = s1x - s0x,    dsy = s1y - s0y;
        sS0x[tid] = s0x;  sS0y[tid] = s0y;
        sDsx[tid] = dsx;  sDsy[tid] = dsy;
        sNs0[tid] = s0x * s0x + s0y * s0y;
        sNds[tid] = dsx * dsx + dsy * dsy;
        sCs[tid]  = dsx * s0x + dsy * s0y;
    }
    __syncthreads();

    // A frag (16x4 f32): row m = [s0x, s0y, dsx, dsy], K=0..3.
    // VGPR0: lanes0-15 K=0 (s0x), lanes16-31 K=2 (dsx); VGPR1: K=1 (s0y) / K=3 (dsy)
    v2f afrag;
    afrag.x = hi ? sDsx[mn] : sS0x[mn];
    afrag.y = hi ? sDsy[mn] : sS0y[mn];

    // Hoist the 8 row stats this lane needs (M = v + 8*hi) out of the sweep:
    // removes all LDS traffic / dscnt waits from the steady-state loop.
    float rNs0[8], rNds[8], rCs[8];
    #pragma unroll
    for (int v = 0; v < 8; ++v) {
        int M = v + (hi << 3);
        rNs0[v] = sNs0[M];
        rNds[v] = sNds[M];
        rCs[v]  = sCs[M];
    }

    const v8f cz = {0.f, 0.f, 0.f, 0.f, 0.f, 0.f, 0.f, 0.f};
    float acc = 0.f;
    for (int jt = wave; jt < (N_NODES_C / 16); jt += 8) {
        int node = jt * 16 + mn;
        const float* zp = Z + (long)node * ZSTRIDE;
        float r0x = zp[k],        r1x = zp[k + 1];
        float r0y = zp[NPK + k],  r1y = zp[NPK + k + 1];
        float drx = r1x - r0x,    dry = r1y - r0y;
        float nr0 = r0x * r0x + r0y * r0y;
        float ndr = drx * drx + dry * dry;
        float cr  = drx * r0x + dry * r0y;

        // B frags (4x16 f32): VGPR0 lanes0-15 = row K=0, lanes16-31 = row K=2;
        //                     VGPR1 lanes0-15 = row K=1, lanes16-31 = row K=3.
        v2f bD, bS, bC;
        bD.x = hi ? 0.f : r0x;   bD.y = hi ? 0.f : r0y;  // -> Zs0 . Zr0
        bS.x = hi ? drx : 0.f;   bS.y = hi ? dry : 0.f;  // -> DZs . DZr
        bC.x = hi ? r0x : drx;   bC.y = hi ? r0y : dry;  // -> Zs0.DZr + DZs.Zr0

        v8f xD = __builtin_amdgcn_wmma_f32_16x16x4_f32(false, afrag, false, bD, (short)0, cz, false, false);
        v8f xS = __builtin_amdgcn_wmma_f32_16x16x4_f32(false, afrag, false, bS, (short)0, cz, false, false);
        v8f xC = __builtin_amdgcn_wmma_f32_16x16x4_f32(false, afrag, false, bC, (short)0, cz, false, false);

        #pragma unroll
        for (int v = 0; v < 8; ++v) {
            float D = rNs0[v] + nr0 - 2.f * xD[v];
            float S = rNds[v] + ndr - 2.f * xS[v];
            float C = xC[v] - rCs[v] - cr;
            // branchless mask (reference: Ss = where(mask, S, 1))
            bool  msk = (D > 0.f) & (S > 0.f);
            float Ss  = msk ? S : 1.0f;
            float mu  = C * __builtin_amdgcn_rcpf(Ss);       // C / Ss
            float s2  = Ss + Ss;
            float sq  = __builtin_amdgcn_sqrtf(s2);          // 1/sigma = sqrt(2S)
            float sig = __builtin_amdgcn_rsqf(s2);           // sigma  = sqrt(0.5/S)
            float e   = fexp(fmaf(C, mu, -D));               // exp(C^2/S - D)
            float val = e * sig * (fnormcdf((1.f - mu) * sq) - fnormcdf(-mu * sq));
            acc += msk ? val : 0.f;
        }
    }

    float s = clpm_block_reduce(acc, red);
    if (tid == 0) {
        float dt = cp[k + 1] - cp[k];
        atomicAdd(ws + 2, 0.5f * dt * s);
    }
}

// ---------------- finalize ----------------
__global__ void clpm_finalize_kernel(const float* ws, const float* beta_p,
                                     const int* fs_p, float bs_f, float e_f,
                                     float* out) {
    float prior    = PENALTY_F * ws[0];
    float fltsum   = ws[1];
    float integral = ws[2];           // includes 0.5 * dt
    float beta     = beta_p[0];
    float scale    = (float)fs_p[0] / bs_f;
    const float sqrt2pi = 2.5066282746310002f;
    out[0] = scale * (prior - beta * e_f - fltsum + sqrt2pi * expf(beta) * integral);
}

extern "C" void kernel_launch(void* const* d_in, const int* in_sizes, int n_in,
                              void* d_out, int out_size, void* d_ws, size_t ws_size,
                              hipStream_t stream) {
    const float* Z     = (const float*)d_in[0];
    const float* beta  = (const float*)d_in[1];
    const float* ts    = (const float*)d_in[2];
    const int*   snd   = (const int*)d_in[3];
    const int*   rcv   = (const int*)d_in[4];
    const int*   nodes = (const int*)d_in[5];
    const int*   su    = (const int*)d_in[6];
    const float* cp    = (const float*)d_in[7];
    const int*   fs    = (const int*)d_in[8];
    float* out = (float*)d_out;
    float* ws  = (float*)d_ws;

    int E  = in_sizes[2];
    int BS = in_sizes[5];

    hipLaunchKernelGGL(clpm_zero_ws, dim3(1), dim3(1), 0, stream, ws);
    hipLaunchKernelGGL(clpm_prior_kernel, dim3((BS + 255) / 256), dim3(256), 0, stream,
                       Z, nodes, BS, ws);
    hipLaunchKernelGGL(clpm_flt_kernel, dim3(256), dim3(256), 0, stream,
                       Z, ts, snd, rcv, cp, E, ws);
    hipLaunchKernelGGL(clpm_integral_kernel, dim3(8 * 128), dim3(256), 0, stream,
                       Z, su, cp, ws);
    hipLaunchKernelGGL(clpm_finalize_kernel, dim3(1), dim3(1), 0, stream,
                       ws, beta, fs, (float)BS, (float)E, out);
}